// D2GroupConvolutionLayer_54322746360397
// MI455X (gfx1250) — compile-verified
//
#include <hip/hip_runtime.h>
#include <hip/hip_bf16.h>

typedef __attribute__((ext_vector_type(2))) float v2f;
typedef __attribute__((ext_vector_type(8))) float v8f;

constexpr int BB = 2;      // batch
constexpr int VV = 4;      // views
constexpr int NN = 2048;   // nodes
constexpr int FF = 128;    // features
constexpr int EE = 32768;  // explicit edges
constexpr int NE = 34816;  // edges + self loops
constexpr int OO = 512;    // out features

// ---------------------------------------------------------------- helpers
__device__ __forceinline__ float sgnval(int r, int j) {
  if (j >= 4) return 1.0f;
  const bool flip = (j & 1) ? ((r & 2) != 0) : ((r & 1) != 0);
  return flip ? -1.0f : 1.0f;
}

__device__ __forceinline__ int src_of(const int* __restrict__ ei, int e) {
  return (e < EE) ? ei[e] : (e - EE);
}
__device__ __forceinline__ int dst_of(const int* __restrict__ ei, int e) {
  return (e < EE) ? ei[EE + e] : (e - EE);
}

// ---------------------------------------------------------------- zero fills
__global__ void zero_f32_kernel(float* __restrict__ p, size_t n) {
  size_t i = (size_t)blockIdx.x * blockDim.x + threadIdx.x;
  if (i < n) p[i] = 0.0f;
}
__global__ void zero_i32_kernel(int* __restrict__ p, int n) {
  int i = blockIdx.x * blockDim.x + threadIdx.x;
  if (i < n) p[i] = 0;
}

// ---------------------------------------------------------------- CSR build
__global__ void count_dst_kernel(const int* __restrict__ ei, int* __restrict__ counts) {
  int e = blockIdx.x * blockDim.x + threadIdx.x;
  if (e >= NE) return;
  atomicAdd(&counts[dst_of(ei, e)], 1);
}

__global__ __launch_bounds__(256) void scan_counts_kernel(const int* __restrict__ counts,
                                                          int* __restrict__ rowstart) {
  __shared__ int part[256];
  __shared__ int pref[257];
  const int tid = threadIdx.x;
  const int base = tid * 8;          // NN == 256*8
  int loc[8];
  int s = 0;
#pragma unroll
  for (int i = 0; i < 8; ++i) { loc[i] = s; s += counts[base + i]; }
  part[tid] = s;
  __syncthreads();
  if (tid == 0) {
    int a = 0;
    for (int i = 0; i < 256; ++i) { pref[i] = a; a += part[i]; }
    pref[256] = a;
  }
  __syncthreads();
  const int off = pref[tid];
#pragma unroll
  for (int i = 0; i < 8; ++i) rowstart[base + i] = off + loc[i];
  if (tid == 255) rowstart[NN] = pref[256];
}

// one wave per destination node: deterministic stable compaction via ballot
__global__ __launch_bounds__(256) void build_rows_kernel(const int* __restrict__ ei,
                                                         const int* __restrict__ rowstart,
                                                         int* __restrict__ order) {
  const int lane = threadIdx.x & 31;
  const int wave = threadIdx.x >> 5;
  const int n = blockIdx.x * 8 + wave;
  if (n >= NN) return;
  const int base = rowstart[n];
  int cnt = 0;
  for (int e0 = 0; e0 < NE; e0 += 32) {
    const int e = e0 + lane;
    const bool match = (e < NE) && (dst_of(ei, e) == n);
    unsigned long long mask = __ballot(match);
    const int rank = __popcll(mask & ((1ull << lane) - 1ull));
    if (match) order[base + cnt + rank] = e;
    cnt += __popcll(mask);
  }
}

// ---------------------------------------------------------------- fp32 WMMA GEMM
// One K-quad: 8 B loads, 8 WMMAs covering a 32x64 wave tile.
__device__ __forceinline__ void wmma_quad(const float* __restrict__ W, int k, int khalf,
                                          int col, v2f a0, v2f a1, v8f* acc) {
  const float* w0 = W + (size_t)(k + khalf) * OO;
  const float* w1 = w0 + OO;
  const v2f b0 = { w0[col +  0], w1[col +  0] };
  const v2f b1 = { w0[col + 16], w1[col + 16] };
  const v2f b2 = { w0[col + 32], w1[col + 32] };
  const v2f b3 = { w0[col + 48], w1[col + 48] };
  acc[0] = __builtin_amdgcn_wmma_f32_16x16x4_f32(false, a0, false, b0, (short)0, acc[0], false, false);
  acc[1] = __builtin_amdgcn_wmma_f32_16x16x4_f32(false, a0, false, b1, (short)0, acc[1], false, false);
  acc[2] = __builtin_amdgcn_wmma_f32_16x16x4_f32(false, a0, false, b2, (short)0, acc[2], false, false);
  acc[3] = __builtin_amdgcn_wmma_f32_16x16x4_f32(false, a0, false, b3, (short)0, acc[3], false, false);
  acc[4] = __builtin_amdgcn_wmma_f32_16x16x4_f32(false, a1, false, b0, (short)0, acc[4], false, false);
  acc[5] = __builtin_amdgcn_wmma_f32_16x16x4_f32(false, a1, false, b1, (short)0, acc[5], false, false);
  acc[6] = __builtin_amdgcn_wmma_f32_16x16x4_f32(false, a1, false, b2, (short)0, acc[6], false, false);
  acc[7] = __builtin_amdgcn_wmma_f32_16x16x4_f32(false, a1, false, b3, (short)0, acc[7], false, false);
}

// H[b,n,:] = (x[b,v1,n,:]*s[v2]) @ W[0:128,:] + (x[b,v2,n,:]*s[v1]) @ W[128:256,:]
// block = 256 threads (8 waves); block tile = 32 rows x 512 cols; wave tile 32x64.
__global__ __launch_bounds__(256) void gemm_pair_kernel(const float* __restrict__ x,
                                                        const float* __restrict__ W,
                                                        float* __restrict__ H,
                                                        int v1, int v2) {
  const int lane = threadIdx.x & 31;
  const int wave = threadIdx.x >> 5;
  const int rowTile = blockIdx.x * 32;          // global row = b*NN + n
  const int b = rowTile >> 11;                  // / NN
  const int nbase = rowTile & (NN - 1);
  const int r0 = nbase + (lane & 15);
  const float* x1a = x + (((size_t)b * VV + v1) * NN + r0) * FF;        // A frag rows 0..15
  const float* x1b = x1a + 16 * FF;                                     // A frag rows 16..31
  const float* x2a = x + (((size_t)b * VV + v2) * NN + r0) * FF;
  const float* x2b = x2a + 16 * FF;
  const int khalf = (lane >> 4) * 2;            // lanes 16-31 hold K+2,K+3
  const int col = wave * 64 + (lane & 15);

  v8f acc[8] = {};

  // k-quad 0: signed (features 0..3 of view-v1 input, sign from view v2)
  {
    v2f a0 = *(const v2f*)(x1a + khalf);
    v2f a1 = *(const v2f*)(x1b + khalf);
    const float s0 = sgnval(v2, khalf);
    const float s1 = sgnval(v2, khalf + 1);
    a0.x *= s0; a0.y *= s1;
    a1.x *= s0; a1.y *= s1;
    wmma_quad(W, 0, khalf, col, a0, a1, acc);
  }
  // k = 4..127: clean
  for (int k = 4; k < FF; k += 4) {
    const v2f a0 = *(const v2f*)(x1a + k + khalf);
    const v2f a1 = *(const v2f*)(x1b + k + khalf);
    wmma_quad(W, k, khalf, col, a0, a1, acc);
  }
  // k-quad 128: signed (features 0..3 of view-v2 input, sign from view v1)
  {
    v2f a0 = *(const v2f*)(x2a + khalf);
    v2f a1 = *(const v2f*)(x2b + khalf);
    const float s0 = sgnval(v1, khalf);
    const float s1 = sgnval(v1, khalf + 1);
    a0.x *= s0; a0.y *= s1;
    a1.x *= s0; a1.y *= s1;
    wmma_quad(W, FF, khalf, col, a0, a1, acc);
  }
  // k = 132..255: clean
  for (int k = FF + 4; k < 2 * FF; k += 4) {
    const v2f a0 = *(const v2f*)(x2a + (k - FF) + khalf);
    const v2f a1 = *(const v2f*)(x2b + (k - FF) + khalf);
    wmma_quad(W, k, khalf, col, a0, a1, acc);
  }

  // D layout: VGPR i -> M = i + 8*(lane>=16), N = lane&15 (per 16-col tile)
  float* Hb = H + (size_t)b * NN * OO;
  const int srow = nbase + 8 * (lane >> 4);
  const int scol = wave * 64 + (lane & 15);
#pragma unroll
  for (int i = 0; i < 8; ++i) {
    float* ra = Hb + (size_t)(srow + i) * OO + scol;
    ra[ 0] = acc[0][i];
    ra[16] = acc[1][i];
    ra[32] = acc[2][i];
    ra[48] = acc[3][i];
    float* rb = Hb + (size_t)(srow + 16 + i) * OO + scol;
    rb[ 0] = acc[4][i];
    rb[16] = acc[5][i];
    rb[32] = acc[6][i];
    rb[48] = acc[7][i];
  }
}

// ---------------------------------------------------------------- per-node attention dots
// u[n] = sum att[0:512]*lrelu(H[n]); v[n] = sum att[512:1024]*lrelu(H[n])
__global__ __launch_bounds__(256) void node_scores_kernel(const float* __restrict__ H,
                                                          const float* __restrict__ att,
                                                          float* __restrict__ uS,
                                                          float* __restrict__ vS) {
  const int lane = threadIdx.x & 31;
  const int wave = threadIdx.x >> 5;
  const int node = blockIdx.x * 8 + wave;        // b*NN + n
  if (node >= BB * NN) return;
  const float* hrow = H + (size_t)node * OO;
  float su = 0.0f, sv = 0.0f;
  for (int j = lane; j < OO; j += 32) {
    const float h = hrow[j];
    const float l = h > 0.0f ? h : 0.2f * h;
    su += att[j] * l;
    sv += att[OO + j] * l;
  }
#pragma unroll
  for (int off = 16; off; off >>= 1) {
    su += __shfl_down(su, off, 32);
    sv += __shfl_down(sv, off, 32);
  }
  if (lane == 0) { uS[node] = su; vS[node] = sv; }
}

// ---------------------------------------------------------------- edge scores
__global__ void edge_scores_kernel(const float* __restrict__ uS, const float* __restrict__ vS,
                                   const int* __restrict__ ei, float* __restrict__ s) {
  const int t = blockIdx.x * blockDim.x + threadIdx.x;
  if (t >= BB * NE) return;
  const int b = t / NE;
  const int e = t - b * NE;
  s[t] = uS[b * NN + src_of(ei, e)] + vS[b * NN + dst_of(ei, e)];
}

// ---------------------------------------------------------------- global softmax stats per batch
__global__ __launch_bounds__(256) void softmax_stats_kernel(const float* __restrict__ s,
                                                            float* __restrict__ stats) {
  const int b = blockIdx.x;
  const int tid = threadIdx.x;
  __shared__ float red[256];
  const float* sb = s + (size_t)b * NE;
  float m = -3.4e38f;
  for (int i = tid; i < NE; i += 256) m = fmaxf(m, sb[i]);
  red[tid] = m;
  __syncthreads();
  for (int off = 128; off; off >>= 1) {
    if (tid < off) red[tid] = fmaxf(red[tid], red[tid + off]);
    __syncthreads();
  }
  const float bm = red[0];
  __syncthreads();
  float sum = 0.0f;
  for (int i = tid; i < NE; i += 256) sum += expf(sb[i] - bm);
  red[tid] = sum;
  __syncthreads();
  for (int off = 128; off; off >>= 1) {
    if (tid < off) red[tid] += red[tid + off];
    __syncthreads();
  }
  if (tid == 0) { stats[2 * b] = bm; stats[2 * b + 1] = red[0]; }
}

// ---------------------------------------------------------------- gather-aggregate into out[:,g]
// out[b,g,n,:] += sum over edges into n of w_e * H[b, src_e, :]
__global__ __launch_bounds__(256) void aggregate_kernel(const float* __restrict__ H,
                                                        const float* __restrict__ s,
                                                        const float* __restrict__ stats,
                                                        const int* __restrict__ rowstart,
                                                        const int* __restrict__ order,
                                                        const int* __restrict__ ei,
                                                        float* __restrict__ out, int g) {
  const int lane = threadIdx.x & 31;
  const int wave = threadIdx.x >> 5;
  const int node = blockIdx.x * 8 + wave;        // b*NN + n
  if (node >= BB * NN) return;
  const int b = node >> 11;
  const int n = node & (NN - 1);
  const float m = stats[2 * b];
  const float inv = 1.0f / stats[2 * b + 1];
  const float* Hb = H + (size_t)b * NN * OO;
  const float* sb = s + (size_t)b * NE;
  float4 acc[4] = {};
  const int beg = rowstart[n];
  const int end = rowstart[n + 1];
  for (int idx = beg; idx < end; ++idx) {
    const int e = order[idx];
    const int sn = src_of(ei, e);
    const float w = expf(sb[e] - m) * inv;
    const float4* hr = (const float4*)(Hb + (size_t)sn * OO);
#pragma unroll
    for (int j = 0; j < 4; ++j) {
      const float4 h4 = hr[lane + 32 * j];
      acc[j].x += w * h4.x;
      acc[j].y += w * h4.y;
      acc[j].z += w * h4.z;
      acc[j].w += w * h4.w;
    }
  }
  float4* op = (float4*)(out + (((size_t)b * VV + g) * NN + n) * OO);
#pragma unroll
  for (int j = 0; j < 4; ++j) {
    float4 o = op[lane + 32 * j];
    o.x += acc[j].x; o.y += acc[j].y; o.z += acc[j].z; o.w += acc[j].w;
    op[lane + 32 * j] = o;
  }
}

// ---------------------------------------------------------------- epilogue: /V + bias
__global__ void epilogue_kernel(float* __restrict__ out, const float* __restrict__ bias, size_t n) {
  size_t i = (size_t)blockIdx.x * blockDim.x + threadIdx.x;
  if (i < n) out[i] = out[i] * 0.25f + bias[i & (OO - 1)];
}

// ---------------------------------------------------------------- host
extern "C" void kernel_launch(void* const* d_in, const int* in_sizes, int n_in,
                              void* d_out, int out_size, void* d_ws, size_t ws_size,
                              hipStream_t stream) {
  const float* x    = (const float*)d_in[0];  // (B,V,N,F)
  const int*   ei   = (const int*)d_in[1];    // (2,E)
  const float* W    = (const float*)d_in[2];  // (2F, O)
  const float* att  = (const float*)d_in[3];  // (1,1,2O)
  const float* bias = (const float*)d_in[4];  // (O)
  float* out = (float*)d_out;                 // (B,V,N,O)

  char* p = (char*)d_ws;
  float* H     = (float*)p; p += (size_t)BB * NN * OO * sizeof(float);   // 8 MB
  float* uS    = (float*)p; p += (size_t)BB * NN * sizeof(float);
  float* vS    = (float*)p; p += (size_t)BB * NN * sizeof(float);
  float* s     = (float*)p; p += (size_t)BB * NE * sizeof(float);
  float* stats = (float*)p; p += 4 * sizeof(float);
  int* counts  = (int*)p;   p += (size_t)NN * sizeof(int);
  int* order   = (int*)p;   p += (size_t)NE * sizeof(int);
  int* rowstart= (int*)p;   p += (size_t)(NN + 1) * sizeof(int);

  const size_t outN = (size_t)BB * VV * NN * OO;

  zero_f32_kernel<<<(unsigned)((outN + 255) / 256), 256, 0, stream>>>(out, outN);
  zero_i32_kernel<<<(NN + 255) / 256, 256, 0, stream>>>(counts, NN);
  count_dst_kernel<<<(NE + 255) / 256, 256, 0, stream>>>(ei, counts);
  scan_counts_kernel<<<1, 256, 0, stream>>>(counts, rowstart);
  build_rows_kernel<<<NN / 8, 256, 0, stream>>>(ei, rowstart, order);

  for (int g = 0; g < VV; ++g) {
    for (int hv = 0; hv < VV; ++hv) {
      const int v1 = hv;
      const int v2 = g ^ hv;
      gemm_pair_kernel<<<(BB * NN) / 32, 256, 0, stream>>>(x, W, H, v1, v2);
      node_scores_kernel<<<(BB * NN) / 8, 256, 0, stream>>>(H, att, uS, vS);
      edge_scores_kernel<<<(BB * NE + 255) / 256, 256, 0, stream>>>(uS, vS, ei, s);
      softmax_stats_kernel<<<BB, 256, 0, stream>>>(s, stats);
      aggregate_kernel<<<(BB * NN) / 8, 256, 0, stream>>>(H, s, stats, rowstart, order, ei, out, g);
    }
  }
  epilogue_kernel<<<(unsigned)((outN + 255) / 256), 256, 0, stream>>>(out, bias, outN);

  (void)in_sizes; (void)n_in; (void)out_size; (void)ws_size;
}